// BilinearInterpolation_51024211476552
// MI455X (gfx1250) — compile-verified
//
#include <hip/hip_runtime.h>

// STN bilinear sampling for MI455X (gfx1250, wave32).
// - einsum('brc,cn') coordinate transform done with V_WMMA_F32_16X16X4_F32
//   (one WMMA computes u,v for a 16-pixel tile per wave).
// - gathers/stores are 128-bit, coalesced; output stores non-temporal so the
//   256 MiB image stays resident in the 192 MB L2 while the 64 MiB output streams.

typedef __attribute__((ext_vector_type(2))) float v2f;
typedef __attribute__((ext_vector_type(4))) float f4;
typedef __attribute__((ext_vector_type(8))) float v8f;

namespace {
constexpr int IB = 32, IH = 128, IW = 512, IC = 32;
constexpr int OH = 64, OW = 256;
constexpr int TILES_PER_ROW   = OW / 16;              // 16 tiles of 16 pixels per output row
constexpr int TILES_PER_BATCH = OH * TILES_PER_ROW;   // 1024
constexpr int TOTAL_WAVES     = IB * TILES_PER_BATCH; // 32768
}

__global__ __launch_bounds__(256) void stn_bilinear_wmma(
    const float* __restrict__ img,    // (32,128,512,32) f32
    const float* __restrict__ theta,  // (32,6) f32
    float* __restrict__ out)          // (32,64,256,32) f32
{
  const int lane  = threadIdx.x & 31;
  const int gwave = blockIdx.x * (blockDim.x >> 5) + (threadIdx.x >> 5);
  const int b  = gwave >> 10;          // batch
  const int nt = gwave & 1023;         // tile within batch
  const int h  = nt >> 4;              // output row 0..63
  const int w0 = (nt & 15) << 4;       // first output col of the 16-pixel tile

  // ---- theta with the reference's debug hack --------------------------------
  float t00 = theta[b * 6 + 0], t01 = theta[b * 6 + 1], t02 = theta[b * 6 + 2];
  float t10 = theta[b * 6 + 3], t11 = theta[b * 6 + 4], t12 = theta[b * 6 + 5];
  if (b == 0) { t10 = 0.0f; t11 = 0.0f; t12 = 0.0f; }   // theta[0,1] = 0
  if (b == 1) { t00 = 0.0f; t01 = 0.0f; t02 = 0.0f; }   // theta[1,0] = 0

  // ---- WMMA coordinate transform: D(16x16) = A(16x4) * B(4x16) --------------
  // A rows 0..1 = theta rows, rows 2..15 = 0.
  // f32 A layout: lanes 0-15 hold K=0 (VGPR0) / K=1 (VGPR1) for row M=lane,
  //               lanes 16-31 hold K=2 / K=3 for row M=lane-16.
  v2f A; A.x = 0.0f; A.y = 0.0f;
  if      (lane == 0)  { A.x = t00; A.y = t01; }
  else if (lane == 1)  { A.x = t10; A.y = t11; }
  else if (lane == 16) { A.x = t02; /* K=3 stays 0 */ }
  else if (lane == 17) { A.x = t12; }

  // B col n = (xg_n, yg_n, 1, 0)^T for the 16 tile pixels (mirrored layout:
  // lanes 0-15 rows K=0/K=1, lanes 16-31 rows K=2/K=3).
  v2f Bm;
  if (lane < 16) {
    Bm.x = __builtin_fmaf((float)(w0 + lane), 2.0f / 255.0f, -1.0f);  // linspace(-1,1,256)
    Bm.y = __builtin_fmaf((float)h,           2.0f / 63.0f,  -1.0f);  // linspace(-1,1,64)
  } else {
    Bm.x = 1.0f;   // homogeneous 1 (K=2)
    Bm.y = 0.0f;   // K=3 pad
  }

  v8f Cz = {};
  // D[0][n] = u_n, D[1][n] = v_n, held in lanes 0-15 (N=lane) of VGPR 0 / 1.
  v8f D = __builtin_amdgcn_wmma_f32_16x16x4_f32(
      /*neg_a=*/false, A, /*neg_b=*/false, Bm,
      /*c_mod=*/(short)0, Cz, /*reuse_a=*/false, /*reuse_b=*/false);

  // ---- gather + blend: 4 pixels x 8 channel-quads per iteration -------------
  const int sub = lane >> 3;         // which of 4 concurrent pixels
  const int cb  = (lane & 7) << 2;   // channel base (0,4,...,28) -> float4
  const f4* img4 = reinterpret_cast<const f4*>(img);
  f4*       out4 = reinterpret_cast<f4*>(out);
  const int obase  = ((b * OH + h) * OW + w0) * IC;
  const int ibbase = b * IH;

  #pragma unroll
  for (int it = 0; it < 4; ++it) {
    const int p = it * 4 + sub;                    // pixel 0..15 within tile
    const float u = __shfl(D[0], p, 32);           // ds_bpermute broadcast
    const float v = __shfl(D[1], p, 32);
    const float x = (u + 1.0f) * 256.0f;           // 0.5*(u+1)*W, W=512
    const float y = (v + 1.0f) * 64.0f;            // 0.5*(v+1)*H, H=128

    int x0 = (int)x, y0 = (int)y;                  // trunc toward zero == astype(int32)
    int x1 = x0 + 1, y1 = y0 + 1;
    x0 = x0 < 0 ? 0 : (x0 > IW - 1 ? IW - 1 : x0);
    x1 = x1 < 0 ? 0 : (x1 > IW - 1 ? IW - 1 : x1);
    y0 = y0 < 0 ? 0 : (y0 > IH - 1 ? IH - 1 : y0);
    y1 = y1 < 0 ? 0 : (y1 > IH - 1 ? IH - 1 : y1);

    const float x0f = (float)x0, x1f = (float)x1;
    const float y0f = (float)y0, y1f = (float)y1;
    const float wA = (x1f - x) * (y1f - y);
    const float wB = (x1f - x) * (y - y0f);
    const float wC = (x - x0f) * (y1f - y);
    const float wD = (x - x0f) * (y - y0f);

    const int rA = (ibbase + y0) * IW;             // row y0 base (pixels)
    const int rB = (ibbase + y1) * IW;             // row y1 base
    const f4 pA = img4[((rA + x0) * IC + cb) >> 2];  // global_load_b128, coalesced
    const f4 pB = img4[((rB + x0) * IC + cb) >> 2];
    const f4 pC = img4[((rA + x1) * IC + cb) >> 2];
    const f4 pD = img4[((rB + x1) * IC + cb) >> 2];

    f4 r;
    r.x = fmaf(pD.x, wD, fmaf(pC.x, wC, fmaf(pB.x, wB, pA.x * wA)));
    r.y = fmaf(pD.y, wD, fmaf(pC.y, wC, fmaf(pB.y, wB, pA.y * wA)));
    r.z = fmaf(pD.z, wD, fmaf(pC.z, wC, fmaf(pB.z, wB, pA.z * wA)));
    r.w = fmaf(pD.w, wD, fmaf(pC.w, wC, fmaf(pB.w, wB, pA.w * wA)));

    // streamed output: non-temporal so the image stays hot in L2
    __builtin_nontemporal_store(r, &out4[(obase + p * IC + cb) >> 2]);
  }
}

extern "C" void kernel_launch(void* const* d_in, const int* in_sizes, int n_in,
                              void* d_out, int out_size, void* d_ws, size_t ws_size,
                              hipStream_t stream) {
  const float* img   = (const float*)d_in[0];   // (32,128,512,32) f32
  const float* theta = (const float*)d_in[1];   // (32,6) f32
  float* out = (float*)d_out;                   // (32,64,256,32) f32

  dim3 block(256);                              // 8 wave32 per block
  dim3 grid(TOTAL_WAVES / 8);                   // 4096 blocks = 32768 waves
  hipLaunchKernelGGL(stn_bilinear_wmma, grid, block, 0, stream, img, theta, out);
}